// HBVMul_33758442946684
// MI455X (gfx1250) — compile-verified
//
#include <hip/hip_runtime.h>
#include <hip/hip_bf16.h>
#include <stdint.h>

#define NSTEP 730
#define NGRID 1500
#define MU    16
#define LENF  15
#define PRECS 1e-5f
#define CH    32            // timesteps per TDM chunk
#define GPB   16            // grids per 256-thread block (16 mu-lanes each)
#define ROWF  (GPB*3)       // 48 floats per forcing row in LDS

typedef __attribute__((ext_vector_type(4))) unsigned int v4u;
typedef __attribute__((ext_vector_type(8))) int          v8i;
typedef __attribute__((ext_vector_type(4))) int          v4i;

// ---------------------------------------------------------------------------
// TDM: 2-D tile load Global->LDS. Descriptor per CDNA5 ISA
// (D# group0: count/lds_addr/global_addr/type=2; group1: data_size=4B, dims,
//  tile dims, dim0 stride). Groups 2/3 + extra group zero (<=2D tensor).
// Tracked with TENSORcnt.
// ---------------------------------------------------------------------------
__device__ __forceinline__ void tdm_load_2d(uint32_t lds_off, uint64_t gaddr,
                                            uint32_t tdim0, uint32_t tdim1,
                                            uint32_t tile0, uint32_t tile1,
                                            uint32_t stride0)
{
    v4u g0;
    v8i g1;
    v4i g2 = (v4i)0, g3 = (v4i)0;
    v8i g4 = (v8i)0;
    g0[0] = 1u;                                             // count=1, user load
    g0[1] = lds_off;                                        // lds_addr (bytes)
    g0[2] = (uint32_t)(gaddr & 0xFFFFFFFFu);                // global_addr[31:0]
    g0[3] = (uint32_t)((gaddr >> 32) & 0x01FFFFFFu)         // global_addr[56:32]
          | (2u << 30);                                     // type = 2 ("image")
    g1[0] = (int)(2u << 16);                                // data_size=2 -> 4B, mask=0
    g1[1] = (int)((tdim0 & 0xFFFFu) << 16);                 // tensor_dim0[15:0]
    g1[2] = (int)(((tdim0 >> 16) & 0xFFFFu) |
                  ((tdim1 & 0xFFFFu) << 16));               // dim0 hi | dim1 lo
    g1[3] = (int)(((tdim1 >> 16) & 0xFFFFu) |
                  ((tile0 & 0xFFFFu) << 16));               // dim1 hi | tile_dim0
    g1[4] = (int)(tile1 & 0xFFFFu);                         // tile_dim1 (tile_dim2=0)
    g1[5] = (int)stride0;                                   // tensor_dim0_stride lo
    g1[6] = 0;                                              // stride0 hi | stride1 lo
    g1[7] = 0;
    __builtin_amdgcn_tensor_load_to_lds(g0, g1, g2, g3, g4, 0);
}

__device__ __forceinline__ float red16(float v) {
    v += __shfl_xor(v, 1, 32);
    v += __shfl_xor(v, 2, 32);
    v += __shfl_xor(v, 4, 32);
    v += __shfl_xor(v, 8, 32);
    return v;                 // sum over the 16-lane mu group (wave32)
}

// ---------------------------------------------------------------------------
// Kernel 1: the sequential HBV scan. One (grid, mu) chain per lane; states in
// VGPRs; forcing double-buffered into LDS by the Tensor Data Mover.
// Writes Qall columns 1..4 (mu-means of Q0,Q1,Q2,ETact) and Qsimave to ws.
// ---------------------------------------------------------------------------
__global__ __launch_bounds__(256) void hbv_scan(
    const float* __restrict__ x,        // (NSTEP, NGRID, 3)
    const float* __restrict__ params,   // (NGRID, 12, MU)
    float* __restrict__ out,            // (NSTEP, NGRID, 5)
    float* __restrict__ qave)           // (NSTEP, NGRID) scratch
{
    __shared__ float sm_x[2][CH * ROWF];

    const int tid     = threadIdx.x;
    const int g_local = tid >> 4;               // 0..15
    const int mu      = tid & 15;
    const int g0      = blockIdx.x * GPB;
    const int g       = g0 + g_local;
    const bool active = (g < NGRID);
    const int  gp     = active ? g : (NGRID - 1);

    // --- unscale parameters (PARA_SCALE baked in) ---
    const float* pb = params + (size_t)gp * 12 * MU + mu;
    const float BETA  =  1.0f   + pb[ 0*MU] * 5.0f;
    const float FC    = 50.0f   + pb[ 1*MU] * 950.0f;
    const float K0    =  0.05f  + pb[ 2*MU] * 0.85f;
    const float K1    =  0.01f  + pb[ 3*MU] * 0.49f;
    const float K2    =  0.001f + pb[ 4*MU] * 0.199f;
    const float LP    =  0.2f   + pb[ 5*MU] * 0.8f;
    const float PERCp =            pb[ 6*MU] * 10.0f;
    const float UZL   =            pb[ 7*MU] * 100.0f;
    const float TT    = -2.5f   + pb[ 8*MU] * 5.0f;
    const float CFMAX =  0.5f   + pb[ 9*MU] * 9.5f;
    const float CFR   =            pb[10*MU] * 0.1f;
    const float CWH   =            pb[11*MU] * 0.2f;
    const float invFC   = 1.0f / FC;
    const float invLPFC = 1.0f / (LP * FC);
    const float CFRX    = CFR * CFMAX;

    float SNOWPACK = 0.001f, MELT = 0.001f, SM = 0.001f, SUZ = 0.001f, SLZ = 0.001f;

    const uint64_t xbase    = (uint64_t)(uintptr_t)x;
    const uint32_t lds_base = (uint32_t)(uintptr_t)(&sm_x[0][0]);
    const uint32_t tdim0    = (uint32_t)(NGRID * 3 - g0 * 3);  // remaining row floats
    const int nchunk = (NSTEP + CH - 1) / CH;

    if (tid < 32) {   // wave 0 drives the TDM (EXEC ignored; one issue per wave)
        tdm_load_2d(lds_base,
                    xbase + 4ull * ((uint64_t)0 * NGRID * 3 + (uint64_t)g0 * 3),
                    tdim0, (uint32_t)NSTEP, ROWF,
                    (uint32_t)(NSTEP < CH ? NSTEP : CH), (uint32_t)(NGRID * 3));
    }

    int t = 0;
    for (int c = 0; c < nchunk; ++c) {
        const int rows = min(CH, NSTEP - c * CH);
        if (tid < 32) {
            if (c + 1 < nchunk) {   // prefetch next chunk into the other buffer
                const int t1 = (c + 1) * CH;
                tdm_load_2d(lds_base + (uint32_t)(((c + 1) & 1) * CH * ROWF * 4),
                            xbase + 4ull * ((uint64_t)t1 * NGRID * 3 + (uint64_t)g0 * 3),
                            tdim0, (uint32_t)(NSTEP - t1), ROWF,
                            (uint32_t)min(CH, NSTEP - t1), (uint32_t)(NGRID * 3));
                __builtin_amdgcn_s_wait_tensorcnt(1);   // current chunk landed
            } else {
                __builtin_amdgcn_s_wait_tensorcnt(0);
            }
        }
        __syncthreads();
        const float* buf = &sm_x[c & 1][0];

        for (int r = 0; r < rows; ++r, ++t) {
            const float Pm = buf[r * ROWF + g_local * 3 + 0];
            const float Tm = buf[r * ROWF + g_local * 3 + 1];
            const float Em = buf[r * ROWF + g_local * 3 + 2];

            const float RAIN = (Tm >= TT) ? Pm : 0.0f;
            const float SNOW = Pm - RAIN;
            SNOWPACK += SNOW;
            const float melt = fminf(fmaxf(CFMAX * (Tm - TT), 0.0f), SNOWPACK);
            MELT     += melt;
            SNOWPACK -= melt;
            const float refrz = fminf(fmaxf(CFRX * (TT - Tm), 0.0f), MELT);
            SNOWPACK += refrz;
            MELT     -= refrz;
            const float tosoil = fmaxf(MELT - CWH * SNOWPACK, 0.0f);
            MELT -= tosoil;

            const float ratio = SM * invFC;                       // > 0 always
            const float sw    = fminf(exp2f(BETA * log2f(ratio)), 1.0f);
            const float rech  = (RAIN + tosoil) * sw;
            SM += RAIN + tosoil - rech;
            const float excess = fmaxf(SM - FC, 0.0f);
            SM -= excess;
            const float evapf = fminf(SM * invLPFC, 1.0f);
            const float ETact = fminf(SM, Em * evapf);
            SM = fmaxf(SM - ETact, PRECS);

            SUZ += rech + excess;
            const float PERCv = fminf(SUZ, PERCp);
            SUZ -= PERCv;
            const float Q0v = K0 * fmaxf(SUZ - UZL, 0.0f);
            SUZ -= Q0v;
            const float Q1v = K1 * SUZ;
            SUZ -= Q1v;
            SLZ += PERCv;
            const float Q2v = K2 * SLZ;
            SLZ -= Q2v;

            // mu-means: 16-lane wave32 shuffle reductions (Qsim mean is linear)
            const float s0 = red16(Q0v);
            const float s1 = red16(Q1v);
            const float s2 = red16(Q2v);
            const float se = red16(ETact);
            if (mu == 0 && active) {
                const float inv = 1.0f / 16.0f;
                float* o = out + ((size_t)t * NGRID + g) * 5;
                o[1] = s0 * inv;
                o[2] = s1 * inv;
                o[3] = s2 * inv;
                o[4] = se * inv;
                qave[(size_t)t * NGRID + g] = (s0 + s1 + s2) * inv;
            }
        }
        __syncthreads();   // buffer may be reused by the next prefetch
    }
}

// ---------------------------------------------------------------------------
// Kernel 2: gamma routing weights, one thread per grid (tiny).
// ---------------------------------------------------------------------------
__global__ __launch_bounds__(256) void hbv_weights(
    const float* __restrict__ rtwts,   // (NGRID, 2)
    float* __restrict__ w)             // (LENF, NGRID)
{
    const int g = blockIdx.x * blockDim.x + threadIdx.x;
    if (g >= NGRID) return;
    const float a  = fmaxf(rtwts[g * 2 + 0] * 2.9f, 0.0f) + 0.1f;
    const float th = fmaxf(rtwts[g * 2 + 1] * 6.5f, 0.0f) + 0.5f;
    const float cc = expf(-lgammaf(a)) * powf(th, -a);
    float wk[LENF];
    float wsum = 0.0f;
    #pragma unroll
    for (int k = 0; k < LENF; ++k) {
        const float tk = (float)k + 0.5f;
        wk[k] = cc * powf(tk, a - 1.0f) * expf(-tk / th);
        wsum += wk[k];
    }
    const float inv = 1.0f / wsum;
    #pragma unroll
    for (int k = 0; k < LENF; ++k) w[k * NGRID + g] = wk[k] * inv;
}

// ---------------------------------------------------------------------------
// Kernel 3: 15-tap causal convolution over time -> Qall column 0.
// Fully parallel over (t, g); coalesced in g; w lives in L2.
// ---------------------------------------------------------------------------
__global__ __launch_bounds__(256) void hbv_route(
    const float* __restrict__ qave,    // (NSTEP, NGRID)
    const float* __restrict__ w,       // (LENF, NGRID)
    float* __restrict__ out)           // (NSTEP, NGRID, 5)
{
    const int idx = blockIdx.x * blockDim.x + threadIdx.x;
    if (idx >= NSTEP * NGRID) return;
    const int t = idx / NGRID;
    const int g = idx - t * NGRID;
    float acc = 0.0f;
    #pragma unroll
    for (int k = 0; k < LENF; ++k) {
        const int ts = t - k;
        if (ts >= 0) acc = fmaf(w[k * NGRID + g], qave[(size_t)ts * NGRID + g], acc);
    }
    out[(size_t)idx * 5 + 0] = acc;
}

extern "C" void kernel_launch(void* const* d_in, const int* in_sizes, int n_in,
                              void* d_out, int out_size, void* d_ws, size_t ws_size,
                              hipStream_t stream)
{
    const float* x      = (const float*)d_in[0];   // (730,1500,3)
    const float* params = (const float*)d_in[1];   // (1500,12,16)
    const float* rtwts  = (const float*)d_in[2];   // (1500,2)
    // d_in[3] = mu scalar (fixed at 16, baked into the kernels)

    float* out  = (float*)d_out;                   // (730,1500,5)
    float* qave = (float*)d_ws;                    // 730*1500 floats
    float* w    = qave + (size_t)NSTEP * NGRID;    // 15*1500 floats

    hbv_scan   <<<dim3((NGRID + GPB - 1) / GPB), dim3(256), 0, stream>>>(x, params, out, qave);
    hbv_weights<<<dim3((NGRID + 255) / 256),     dim3(256), 0, stream>>>(rtwts, w);
    hbv_route  <<<dim3((NSTEP * NGRID + 255) / 256), dim3(256), 0, stream>>>(qave, w, out);
}